// Encoder_66795331387595
// MI455X (gfx1250) — compile-verified
//
#include <hip/hip_runtime.h>
#include <math.h>

typedef float v2f __attribute__((ext_vector_type(2)));
typedef float v8f __attribute__((ext_vector_type(8)));

#define N_NODES   4096
#define NFEAT     256
#define NHID      128
#define N_EDGES   131072
#define KTOP      64
#define MASK_WORDS 128   // 4096 bits / 32

#define NEG_BIG (-3.402823466e+38f)

// ---------------------------------------------------------------------------
// Adjacency build: bitmask rows (4096 x 128 u32), self-loops included.
// ---------------------------------------------------------------------------
__global__ void init_mask_kernel(unsigned* __restrict__ mask) {
    int idx = blockIdx.x * blockDim.x + threadIdx.x;   // 0 .. 524287
    int row = idx >> 7;      // / MASK_WORDS
    int w   = idx & 127;
    unsigned v = 0u;
    if ((row >> 5) == w) v = 1u << (row & 31);         // self-loop bit
    mask[idx] = v;
}

__global__ void scatter_edges_kernel(const int* __restrict__ ei,
                                     unsigned* __restrict__ mask) {
    int e = blockIdx.x * blockDim.x + threadIdx.x;
    if (e < N_EDGES) {
        int r = ei[e];             // edge_index[0, e]
        int c = ei[N_EDGES + e];   // edge_index[1, e]
        atomicOr(&mask[r * MASK_WORDS + (c >> 5)], 1u << (c & 31));
    }
}

// top_k of a binary row with stable ties == first min(deg,64) set bits ascending
__global__ void extract_nbrs_kernel(const unsigned* __restrict__ mask,
                                    int* __restrict__ nbr,
                                    int* __restrict__ deg) {
    int row = blockIdx.x * blockDim.x + threadIdx.x;
    if (row >= N_NODES) return;
    int cnt = 0, full = 0;
    for (int w = 0; w < MASK_WORDS; ++w) {
        unsigned bits = mask[row * MASK_WORDS + w];
        full += __popc(bits);
        while (bits) {
            int b = __ffs(bits) - 1;
            bits &= bits - 1u;
            if (cnt < KTOP) nbr[row * KTOP + cnt] = w * 32 + b;
            cnt++;
        }
    }
    deg[row] = full;   // row_sum (includes self loop)
}

// ---------------------------------------------------------------------------
// C[M,NHID] = A[M,K] @ B[K,NHID] via V_WMMA_F32_16X16X4_F32.
// One wave per 16x16 output tile; exact grid coverage (EXEC all ones).
// ---------------------------------------------------------------------------
__global__ void gemm_wmma_kernel(const float* __restrict__ A,
                                 const float* __restrict__ B,
                                 float* __restrict__ C,
                                 int K, int ntilesN) {
    int wave = (blockIdx.x * blockDim.x + threadIdx.x) >> 5;
    int lane = threadIdx.x & 31;
    int tI = wave / ntilesN;
    int tJ = wave - tI * ntilesN;
    int idx = lane & 15;
    int kf  = (lane < 16) ? 0 : 2;
    const float* arow = A + (size_t)(tI * 16 + idx) * K;
    v8f c = {};
    for (int k = 0; k < K; k += 4) {
        v2f a, b;
        a.x = arow[k + kf];
        a.y = arow[k + kf + 1];
        b.x = B[(size_t)(k + kf)     * NHID + tJ * 16 + idx];
        b.y = B[(size_t)(k + kf + 1) * NHID + tJ * 16 + idx];
        c = __builtin_amdgcn_wmma_f32_16x16x4_f32(false, a, false, b,
                                                  (short)0, c, false, false);
    }
    int rbase = tI * 16 + ((lane < 16) ? 0 : 8);
    int col   = tJ * 16 + idx;
#pragma unroll
    for (int r = 0; r < 8; ++r)
        C[(size_t)(rbase + r) * NHID + col] = c[r];
}

__global__ void row_sq_kernel(const float* __restrict__ z,
                              float* __restrict__ sq) {
    int row = blockIdx.x * blockDim.x + threadIdx.x;
    if (row >= N_NODES) return;
    const float* p = z + (size_t)row * NHID;
    float s = 0.f;
    for (int d = 0; d < NHID; ++d) s += p[d] * p[d];
    sq[row] = s;
}

// ---------------------------------------------------------------------------
// Gram via fp32 WMMA, fused into l2[i,j] = sqrt(max(sq_i + sq_j - 2*z_i.z_j, 0))
// (exactly matches _pairwise_l2's safe-sqrt semantics).
// ---------------------------------------------------------------------------
__global__ void gram_l2_kernel(const float* __restrict__ z,
                               const float* __restrict__ sq,
                               float* __restrict__ l2) {
    int wave = (blockIdx.x * blockDim.x + threadIdx.x) >> 5;
    int lane = threadIdx.x & 31;
    int tI = wave >> 8;      // / 256 tiles
    int tJ = wave & 255;
    int idx = lane & 15;
    int kf  = (lane < 16) ? 0 : 2;
    const float* za = z + (size_t)(tI * 16 + idx) * NHID;
    const float* zb = z + (size_t)(tJ * 16 + idx) * NHID;
    v8f c = {};
    for (int k = 0; k < NHID; k += 4) {
        v2f a, b;
        a.x = za[k + kf]; a.y = za[k + kf + 1];
        b.x = zb[k + kf]; b.y = zb[k + kf + 1];
        c = __builtin_amdgcn_wmma_f32_16x16x4_f32(false, a, false, b,
                                                  (short)0, c, false, false);
    }
    int rbase = tI * 16 + ((lane < 16) ? 0 : 8);
    int col   = tJ * 16 + idx;
    float sqc = sq[col];
#pragma unroll
    for (int r = 0; r < 8; ++r) {
        float d2 = sq[rbase + r] + sqc - 2.f * c[r];
        l2[(size_t)(rbase + r) * N_NODES + col] = (d2 > 0.f) ? sqrtf(d2) : 0.f;
    }
}

// ---------------------------------------------------------------------------
// Soft weighted medoid aggregation + bias + relu. One block (128 thr) per row.
// dist[k] = sum_{j in N(i)} l2[nbr_j, nbr_k]; softmax(-dist/(0.25*row_sum));
// out[i,:] = relu(row_sum * sum_k w_k * z[nbr_k,:] + bias).
// ---------------------------------------------------------------------------
__global__ void aggregate_kernel(const float* __restrict__ z,
                                 const float* __restrict__ l2,
                                 const int* __restrict__ nbr,
                                 const int* __restrict__ deg,
                                 const float* __restrict__ bias,
                                 float* __restrict__ out) {
    int row = blockIdx.x;
    int t   = threadIdx.x;   // 0..127
    __shared__ int   s_nbr[KTOP];
    __shared__ float s_w[KTOP];
    __shared__ float s_red[KTOP];

    int full = deg[row];
    int nd   = full < KTOP ? full : KTOP;
    float rs = (float)full;

    if (t < KTOP) s_nbr[t] = (t < nd) ? nbr[row * KTOP + t] : 0;
    __syncthreads();

    float logit = NEG_BIG;
    if (t < nd) {
        int m = s_nbr[t];
        float dist = 0.f;
        for (int j = 0; j < nd; ++j)
            dist += l2[(size_t)s_nbr[j] * N_NODES + m];
        logit = -dist / (0.25f * rs);
    }
    if (t < KTOP) s_red[t] = logit;
    __syncthreads();
    for (int s = 32; s > 0; s >>= 1) {
        if (t < s) s_red[t] = fmaxf(s_red[t], s_red[t + s]);
        __syncthreads();
    }
    float mx = s_red[0];
    __syncthreads();

    float w = (t < nd) ? __expf(logit - mx) : 0.f;
    if (t < KTOP) s_red[t] = w;
    __syncthreads();
    for (int s = 32; s > 0; s >>= 1) {
        if (t < s) s_red[t] += s_red[t + s];
        __syncthreads();
    }
    float sum = s_red[0];
    if (t < KTOP) s_w[t] = w / sum;
    __syncthreads();

    float acc = 0.f;
    for (int k = 0; k < nd; ++k)
        acc += s_w[k] * z[(size_t)s_nbr[k] * NHID + t];
    float o = rs * acc + bias[t];
    out[(size_t)row * NHID + t] = o > 0.f ? o : 0.f;
}

// ---------------------------------------------------------------------------
extern "C" void kernel_launch(void* const* d_in, const int* in_sizes, int n_in,
                              void* d_out, int out_size, void* d_ws, size_t ws_size,
                              hipStream_t stream) {
    const float* x  = (const float*)d_in[0];   // [4096,256]
    const int*   ei = (const int*)  d_in[1];   // [2,131072]
    const float* W1 = (const float*)d_in[2];   // [256,128]
    const float* b1 = (const float*)d_in[3];   // [128]
    const float* W2 = (const float*)d_in[4];   // [128,128]
    const float* b2 = (const float*)d_in[5];   // [128]
    float* out = (float*)d_out;                // [4096,128]

    char* ws = (char*)d_ws;
    size_t off = 0;
    float*    l2   = (float*)   (ws + off); off += (size_t)N_NODES * N_NODES * 4; // 64MB
    float*    z    = (float*)   (ws + off); off += (size_t)N_NODES * NHID * 4;    // 2MB
    float*    h    = (float*)   (ws + off); off += (size_t)N_NODES * NHID * 4;    // 2MB
    unsigned* mask = (unsigned*)(ws + off); off += (size_t)N_NODES * MASK_WORDS * 4; // 2MB
    int*      nbr  = (int*)     (ws + off); off += (size_t)N_NODES * KTOP * 4;    // 1MB
    int*      deg  = (int*)     (ws + off); off += (size_t)N_NODES * 4;
    float*    sq   = (float*)   (ws + off); off += (size_t)N_NODES * 4;

    // Adjacency build
    init_mask_kernel   <<<2048, 256, 0, stream>>>(mask);
    scatter_edges_kernel<<<512, 256, 0, stream>>>(ei, mask);
    extract_nbrs_kernel<<<16,  256, 0, stream>>>(mask, nbr, deg);

    // Layer 1: z = x @ W1 ; h = relu(SWM(z) + b1)
    gemm_wmma_kernel<<<256, 256, 0, stream>>>(x, W1, z, NFEAT, 8);   // 2048 waves
    row_sq_kernel   <<<16,  256, 0, stream>>>(z, sq);
    gram_l2_kernel  <<<8192,256, 0, stream>>>(z, sq, l2);            // 65536 waves
    aggregate_kernel<<<N_NODES, 128, 0, stream>>>(z, l2, nbr, deg, b1, h);

    // Layer 2: z = h @ W2 ; out = relu(SWM(z) + b2)
    gemm_wmma_kernel<<<256, 256, 0, stream>>>(h, W2, z, NHID, 8);
    row_sq_kernel   <<<16,  256, 0, stream>>>(z, sq);
    gram_l2_kernel  <<<8192,256, 0, stream>>>(z, sq, l2);
    aggregate_kernel<<<N_NODES, 128, 0, stream>>>(z, l2, nbr, deg, b2, out);
}